// scMGCNLayer_56882546868390
// MI455X (gfx1250) — compile-verified
//
#include <hip/hip_runtime.h>
#include <hip/hip_bf16.h>
#include <cstddef>

#define HDIM 128
#define GNUM 3
#define INDIM 512

typedef float v2f __attribute__((ext_vector_type(2)));
typedef float v8f __attribute__((ext_vector_type(8)));

// ---------------------------------------------------------------------------
// Degree accumulation: outd[src[e]] += 1, ind[dst[e]] += 1  (hardware f32 atomics)
// ---------------------------------------------------------------------------
__global__ __launch_bounds__(256) void degree_kernel(
    const int* __restrict__ src, const int* __restrict__ dst,
    float* __restrict__ outd, float* __restrict__ ind, int E) {
  int e = blockIdx.x * blockDim.x + threadIdx.x;
  if (e < E) {
    unsafeAtomicAdd(&outd[src[e]], 1.0f);
    unsafeAtomicAdd(&ind[dst[e]], 1.0f);
  }
}

// In-place: deg -> clip(deg,1)^-0.5
__global__ __launch_bounds__(256) void norm_kernel(
    float* __restrict__ ns, float* __restrict__ nd, int n) {
  int i = blockIdx.x * blockDim.x + threadIdx.x;
  if (i < n) {
    ns[i] = rsqrtf(fmaxf(ns[i], 1.0f));
    nd[i] = rsqrtf(fmaxf(nd[i], 1.0f));
  }
}

// ---------------------------------------------------------------------------
// C[nrows x 128] = A[nrows x K] @ W[K x 128] using V_WMMA_F32_16X16X4_F32.
// Block = 256 threads (8 waves). Block handles one 16-row M tile; wave w
// produces the 16x16 tile at columns [16w, 16w+16). A tile staged in LDS.
// ---------------------------------------------------------------------------
__global__ __launch_bounds__(256) void gemm_wmma_kernel(
    const float* __restrict__ A, const float* __restrict__ W,
    float* __restrict__ C, int nrows, int K) {
  __shared__ float As[16 * INDIM];  // 32 KB, covers K up to 512

  const int row0 = blockIdx.x * 16;
  const int tid  = threadIdx.x;

  // Cooperative, K-contiguous load of the 16 x K A tile into LDS.
  for (int r = 0; r < 16; ++r) {
    int gr = row0 + r;
    const float* arow = A + (size_t)gr * K;
    for (int c = tid; c < K; c += 256) {
      As[r * K + c] = (gr < nrows) ? arow[c] : 0.0f;
    }
  }
  __syncthreads();

  const int wave = tid >> 5;
  const int lane = tid & 31;
  const int half = lane >> 4;   // 0: lanes 0-15, 1: lanes 16-31
  const int idx  = lane & 15;
  const int col0 = wave * 16;

  v8f acc = {};
  for (int kk = 0; kk < K; kk += 4) {
    // A fragment: lane holds A[idx][kk + 2*half + {0,1}]
    v2f a, b;
    a.x = As[idx * K + kk + 2 * half];
    a.y = As[idx * K + kk + 2 * half + 1];
    // B fragment: lane holds W[kk + 2*half + {0,1}][col0 + idx]
    b.x = W[(size_t)(kk + 2 * half) * HDIM + col0 + idx];
    b.y = W[(size_t)(kk + 2 * half + 1) * HDIM + col0 + idx];
    acc = __builtin_amdgcn_wmma_f32_16x16x4_f32(
        false, a, false, b, (short)0, acc, false, false);
  }

  // D layout: acc[j] = C[row0 + j + 8*half][col0 + idx]
  for (int j = 0; j < 8; ++j) {
    int r = row0 + j + 8 * half;
    if (r < nrows) C[(size_t)r * HDIM + col0 + idx] = acc[j];
  }
}

// ---------------------------------------------------------------------------
// Edge scatter: agg[dst[e]*stride + j] += x[src[e]*128 + j] * ns[src[e]]
// One wave per edge: 32 lanes x float4 = 128 floats (512B coalesced row).
// ---------------------------------------------------------------------------
__global__ __launch_bounds__(256) void scatter_kernel(
    const float* __restrict__ x, const float* __restrict__ ns,
    const int* __restrict__ src, const int* __restrict__ dst,
    float* __restrict__ agg, int rowStride, int E) {
  const int nwarps = gridDim.x * (blockDim.x >> 5);
  const int warp   = blockIdx.x * (blockDim.x >> 5) + (threadIdx.x >> 5);
  const int lane   = threadIdx.x & 31;

  for (int e = warp; e < E; e += nwarps) {
    const int s = src[e];
    const int d = dst[e];
    const float scale = ns[s];
    const float4 v = ((const float4*)(x + (size_t)s * HDIM))[lane];
    float* ar = agg + (size_t)d * rowStride + lane * 4;
    unsafeAtomicAdd(&ar[0], v.x * scale);
    unsafeAtomicAdd(&ar[1], v.y * scale);
    unsafeAtomicAdd(&ar[2], v.z * scale);
    unsafeAtomicAdd(&ar[3], v.w * scale);
  }
}

// out = agg * nd[node] + bias (+ optional ELU). Strided in/out (may alias).
__global__ __launch_bounds__(256) void finalize_kernel(
    const float* __restrict__ agg, int aggStride,
    const float* __restrict__ nd, const float* __restrict__ bias,
    float* __restrict__ out, int outStride, int n, int act) {
  int i = blockIdx.x * blockDim.x + threadIdx.x;
  if (i >= n * HDIM) return;
  int node = i >> 7;
  int j    = i & (HDIM - 1);
  float v = agg[(size_t)node * aggStride + j] * nd[node] + bias[j];
  if (act) v = (v > 0.0f) ? v : (__expf(v) - 1.0f);
  out[(size_t)node * outStride + j] = v;
}

// ---------------------------------------------------------------------------
// Attention score accumulation: for row r = n*G+g of z[N*G,128]:
//   w = sum_j tanh(z_r . Wa1[:,j] + ba1[j]) * Wa2[j];  wsum[g] += w
// ---------------------------------------------------------------------------
__global__ __launch_bounds__(128) void attn_kernel(
    const float* __restrict__ z, const float* __restrict__ Wa1,
    const float* __restrict__ ba1, const float* __restrict__ Wa2,
    float* __restrict__ wsum) {
  __shared__ float zrow[HDIM];
  __shared__ float red[HDIM];
  const int r = blockIdx.x;
  const int j = threadIdx.x;

  zrow[j] = z[(size_t)r * HDIM + j];
  __syncthreads();

  float acc = ba1[j];
  #pragma unroll 4
  for (int k = 0; k < HDIM; ++k) acc += zrow[k] * Wa1[k * HDIM + j];
  red[j] = tanhf(acc) * Wa2[j];
  __syncthreads();

  for (int s = 64; s > 0; s >>= 1) {
    if (j < s) red[j] += red[j + s];
    __syncthreads();
  }
  if (j == 0) unsafeAtomicAdd(&wsum[r % GNUM], red[0]);
}

// beta = softmax(wsum / N) over the 3 graphs
__global__ void beta_kernel(const float* __restrict__ wsum,
                            float* __restrict__ beta, float invN) {
  if (threadIdx.x == 0 && blockIdx.x == 0) {
    float v0 = wsum[0] * invN, v1 = wsum[1] * invN, v2 = wsum[2] * invN;
    float m  = fmaxf(v0, fmaxf(v1, v2));
    float e0 = __expf(v0 - m), e1 = __expf(v1 - m), e2 = __expf(v2 - m);
    float s  = e0 + e1 + e2;
    beta[0] = e0 / s; beta[1] = e1 / s; beta[2] = e2 / s;
  }
}

// out[n,j] = sum_g beta[g] * z[n,g,j]
__global__ __launch_bounds__(256) void combine_kernel(
    const float* __restrict__ z, const float* __restrict__ beta,
    float* __restrict__ out, int n) {
  int i = blockIdx.x * blockDim.x + threadIdx.x;
  if (i >= n * HDIM) return;
  int node = i >> 7;
  int j    = i & (HDIM - 1);
  const float* zr = z + (size_t)node * GNUM * HDIM + j;
  out[i] = beta[0] * zr[0] + beta[1] * zr[HDIM] + beta[2] * zr[2 * HDIM];
}

// ---------------------------------------------------------------------------
extern "C" void kernel_launch(void* const* d_in, const int* in_sizes, int n_in,
                              void* d_out, int out_size, void* d_ws, size_t ws_size,
                              hipStream_t stream) {
  const float* h   = (const float*)d_in[0];
  const int*   srcs[GNUM] = {(const int*)d_in[1], (const int*)d_in[3], (const int*)d_in[5]};
  const int*   dsts[GNUM] = {(const int*)d_in[2], (const int*)d_in[4], (const int*)d_in[6]};
  const float* W0  = (const float*)d_in[7];
  const float* b0  = (const float*)d_in[8];
  const float* Wh  = (const float*)d_in[9];
  const float* bh  = (const float*)d_in[10];
  const float* Wa1 = (const float*)d_in[11];
  const float* ba1 = (const float*)d_in[12];
  const float* Wa2 = (const float*)d_in[13];
  float* out = (float*)d_out;

  const int N = in_sizes[0] / INDIM;   // 50000
  const int E = in_sizes[1];           // 800000
  const size_t NH = (size_t)N * HDIM;

  // Workspace layout (floats)
  float* bufX   = (float*)d_ws;          // N*H  layer activations
  float* bufW   = bufX   + NH;           // N*H  GEMM output (pre-scatter)
  float* bufAgg = bufW   + NH;           // N*H  scatter accumulator
  float* z      = bufAgg + NH;           // N*G*H
  float* ns     = z      + NH * GNUM;    // N
  float* nd     = ns     + N;            // N
  float* wsum   = nd     + N;            // G
  float* beta   = wsum   + GNUM;         // G

  const int tE   = (E + 255) / 256;
  const int tN   = (N + 255) / 256;
  const int tNH  = (int)((NH + 255) / 256);
  const int mtiles = (N + 15) / 16;
  const int scatterBlocks = 4096;        // 32768 waves, grid-stride over edges

  hipMemsetAsync(z, 0, NH * GNUM * sizeof(float), stream);

  for (int g = 0; g < GNUM; ++g) {
    const int* src = srcs[g];
    const int* dst = dsts[g];

    // Degree norms
    hipMemsetAsync(ns, 0, N * sizeof(float), stream);
    hipMemsetAsync(nd, 0, N * sizeof(float), stream);
    degree_kernel<<<tE, 256, 0, stream>>>(src, dst, ns, nd, E);
    norm_kernel<<<tN, 256, 0, stream>>>(ns, nd, N);

    // Layer 0: h[N,512] @ W0[g] -> scatter -> ELU
    gemm_wmma_kernel<<<mtiles, 256, 0, stream>>>(h, W0 + (size_t)g * INDIM * HDIM, bufW, N, INDIM);
    hipMemsetAsync(bufAgg, 0, NH * sizeof(float), stream);
    scatter_kernel<<<scatterBlocks, 256, 0, stream>>>(bufW, ns, src, dst, bufAgg, HDIM, E);
    finalize_kernel<<<tNH, 256, 0, stream>>>(bufAgg, HDIM, nd, b0 + g * HDIM, bufX, HDIM, N, 1);

    // Layer 1: bufX @ Wh[g,0] -> scatter -> ELU
    gemm_wmma_kernel<<<mtiles, 256, 0, stream>>>(bufX, Wh + (size_t)(g * 2 + 0) * HDIM * HDIM, bufW, N, HDIM);
    hipMemsetAsync(bufAgg, 0, NH * sizeof(float), stream);
    scatter_kernel<<<scatterBlocks, 256, 0, stream>>>(bufW, ns, src, dst, bufAgg, HDIM, E);
    finalize_kernel<<<tNH, 256, 0, stream>>>(bufAgg, HDIM, nd, bh + (g * 2 + 0) * HDIM, bufX, HDIM, N, 1);

    // Layer 2: bufX @ Wh[g,1] -> scatter directly into z[:,g,:] -> no act
    gemm_wmma_kernel<<<mtiles, 256, 0, stream>>>(bufX, Wh + (size_t)(g * 2 + 1) * HDIM * HDIM, bufW, N, HDIM);
    scatter_kernel<<<scatterBlocks, 256, 0, stream>>>(bufW, ns, src, dst, z + g * HDIM, GNUM * HDIM, E);
    finalize_kernel<<<tNH, 256, 0, stream>>>(z + g * HDIM, GNUM * HDIM, nd, bh + (g * 2 + 1) * HDIM,
                                             z + g * HDIM, GNUM * HDIM, N, 0);
  }

  // Attention readout
  hipMemsetAsync(wsum, 0, GNUM * sizeof(float), stream);
  attn_kernel<<<N * GNUM, 128, 0, stream>>>(z, Wa1, ba1, Wa2, wsum);
  beta_kernel<<<1, 1, 0, stream>>>(wsum, beta, 1.0f / (float)N);
  combine_kernel<<<tNH, 256, 0, stream>>>(z, beta, out, N);
}